// MultiBoxLoss_60421599920278
// MI455X (gfx1250) — compile-verified
//
#include <hip/hip_runtime.h>
#include <stdint.h>

#define B_    64
#define P_    24564
#define O_    32
#define C_    5
#define NBLK_ 256
#define VAR0_ 0.1f
#define VAR1_ 0.2f

// ---------------------------------------------------------------------------
// Kernel A: per-batch matching (one workgroup per batch, 256 threads = 8 waves)
// ---------------------------------------------------------------------------
__global__ __launch_bounds__(256)
void mb_match(const float* __restrict__ priors,
              const float* __restrict__ targets,
              uint8_t* __restrict__ conf,
              uint8_t* __restrict__ bti,
              float* __restrict__ wscale)
{
    __shared__ __align__(16) float sh_t[O_ * 5];   // staged raw targets
    __shared__ float tr[O_][4];                    // scaled truths (xyxy)
    __shared__ float ar[O_];                       // truth areas
    __shared__ float lb[O_];                       // labels
    __shared__ float sscale;
    __shared__ unsigned long long red[8];
    __shared__ int bestp[O_];

    const int b   = blockIdx.x;
    const int tid = threadIdx.x;

    // ---- async-stage targets[b] (160 floats = 40 x B128) into LDS ----
    if (tid < 40) {
        unsigned lds = (unsigned)(uintptr_t)(&sh_t[tid * 4]);
        unsigned long long ga =
            (unsigned long long)(uintptr_t)(targets + (size_t)b * O_ * 5 + tid * 4);
        asm volatile("global_load_async_to_lds_b128 %0, %1, off"
                     :: "v"(lds), "v"(ga) : "memory");
    }
    asm volatile("s_wait_asynccnt 0" ::: "memory");
    __syncthreads();

    if (tid == 0) {
        int cnt = 0;
        for (int o = 0; o < O_; ++o) {
            float l = sh_t[o * 5 + 4];
            cnt += (l == 1.0f) ? 1 : 0;
            cnt += (l == 4.0f) ? 1 : 0;
        }
        float s = powf(1.5f, (float)cnt);
        sscale = s;
        wscale[b] = s;
    }
    __syncthreads();
    if (tid < O_) {
        float s  = sscale;
        float x1 = sh_t[tid * 5 + 0], y1 = sh_t[tid * 5 + 1];
        float x2 = sh_t[tid * 5 + 2] * s, y2 = sh_t[tid * 5 + 3] * s;
        tr[tid][0] = x1; tr[tid][1] = y1; tr[tid][2] = x2; tr[tid][3] = y2;
        ar[tid] = (x2 - x1) * (y2 - y1);
        lb[tid] = sh_t[tid * 5 + 4];
    }
    __syncthreads();

    const float4* pr4 = (const float4*)priors;
    const size_t base = (size_t)b * P_;

    // ---- pass 1: per prior, best truth (first-index ties, like jnp.argmax) ----
    for (int p = tid; p < P_; p += 256) {
        float4 pr = pr4[p];
        float bx1 = pr.x - pr.z * 0.5f, by1 = pr.y - pr.w * 0.5f;
        float bx2 = pr.x + pr.z * 0.5f, by2 = pr.y + pr.w * 0.5f;
        float areab = (bx2 - bx1) * (by2 - by1);
        float best = -1.0f; int bo = 0;
        #pragma unroll 8
        for (int o = 0; o < O_; ++o) {
            float lx = fmaxf(tr[o][0], bx1), ly = fmaxf(tr[o][1], by1);
            float rx = fminf(tr[o][2], bx2), ry = fminf(tr[o][3], by2);
            float w  = fmaxf(rx - lx, 0.0f), h = fmaxf(ry - ly, 0.0f);
            float inter = w * h;
            float iou = inter / (ar[o] + areab - inter);
            if (iou > best) { best = iou; bo = o; }
        }
        bti[base + p]  = (uint8_t)bo;
        conf[base + p] = (best < 0.5f) ? (uint8_t)0 : (uint8_t)(int)lb[bo];
    }
    __threadfence_block();

    // ---- pass 2: per truth, best prior (packed u64 max, tie -> smallest p) ----
    const int lane = tid & 31, wv = tid >> 5;
    for (int o = 0; o < O_; ++o) {
        float tx1 = tr[o][0], ty1 = tr[o][1], tx2 = tr[o][2], ty2 = tr[o][3];
        float areaa = ar[o];
        unsigned long long lbest = 0ull;
        for (int p = tid; p < P_; p += 256) {
            float4 pr = pr4[p];
            float bx1 = pr.x - pr.z * 0.5f, by1 = pr.y - pr.w * 0.5f;
            float bx2 = pr.x + pr.z * 0.5f, by2 = pr.y + pr.w * 0.5f;
            float areab = (bx2 - bx1) * (by2 - by1);
            float lx = fmaxf(tx1, bx1), ly = fmaxf(ty1, by1);
            float rx = fminf(tx2, bx2), ry = fminf(ty2, by2);
            float w  = fmaxf(rx - lx, 0.0f), h = fmaxf(ry - ly, 0.0f);
            float inter = w * h;
            float iou = inter / (areaa + areab - inter);
            unsigned long long pk =
                ((unsigned long long)__float_as_uint(iou) << 32) | (unsigned)(~p);
            lbest = (pk > lbest) ? pk : lbest;
        }
        // wave32 reduce
        for (int off = 16; off > 0; off >>= 1) {
            unsigned long long other = __shfl_down(lbest, off);
            lbest = (other > lbest) ? other : lbest;
        }
        if (lane == 0) red[wv] = lbest;
        __syncthreads();
        if (tid == 0) {
            unsigned long long m = red[0];
            for (int w = 1; w < 8; ++w) m = (red[w] > m) ? red[w] : m;
            bestp[o] = (int)(~(unsigned)m);
        }
        __syncthreads();
    }

    // ---- pass 3: forced matches, sequential => last-wins scatter semantics ----
    if (tid == 0) {
        for (int o = 0; o < O_; ++o) {
            int p = bestp[o];
            bti[base + p]  = (uint8_t)o;
            conf[base + p] = (uint8_t)(int)lb[o];   // overlap forced to 2.0 >= 0.5
        }
    }
}

// ---------------------------------------------------------------------------
// Kernel B: OHEM per batch — exact stable-rank selection without a full sort
// ---------------------------------------------------------------------------
__global__ __launch_bounds__(256)
void mb_ohem(const float* __restrict__ conf_data,
             const uint8_t* __restrict__ conf,
             float* __restrict__ lossh,
             uint8_t* __restrict__ neg,
             int* __restrict__ numpos,
             int* __restrict__ numneg)
{
    __shared__ int shc;
    __shared__ int eqp[256];
    const int b = blockIdx.x, tid = threadIdx.x;
    const size_t base = (size_t)b * P_;

    // num_pos
    if (tid == 0) shc = 0;
    __syncthreads();
    {
        int local = 0;
        for (int p = tid; p < P_; p += 256) local += (conf[base + p] > 0) ? 1 : 0;
        atomicAdd(&shc, local);
    }
    __syncthreads();
    const int npos = shc;
    const int K = min(max(3 * npos, 3), P_ - 1);
    __syncthreads();

    // loss_hard = pos ? 0 : lse - x[0]   (conf_t==0 for non-pos)
    for (int p = tid; p < P_; p += 256) {
        int c = conf[base + p];
        float lh = 0.0f;
        if (c == 0) {
            const float* x = conf_data + (base + p) * (size_t)C_;
            float m = x[0];
            #pragma unroll
            for (int j = 1; j < C_; ++j) m = fmaxf(m, x[j]);
            float s = 0.0f;
            #pragma unroll
            for (int j = 0; j < C_; ++j) s += expf(x[j] - m);
            float lse = m + logf(s);
            lh = lse - x[0];
        }
        lossh[base + p] = lh;
    }
    __threadfence_block();
    __syncthreads();

    // binary search for K-th largest value over nonneg float bit space
    unsigned lo = 0u, hi = 0xFFFFFFFFu;
    while (lo < hi) {
        unsigned mid = lo + ((hi - lo) >> 1);
        if (tid == 0) shc = 0;
        __syncthreads();
        int cnt = 0;
        for (int p = tid; p < P_; p += 256)
            cnt += (__float_as_uint(lossh[base + p]) > mid) ? 1 : 0;
        atomicAdd(&shc, cnt);
        __syncthreads();
        int total = shc;
        __syncthreads();
        if (total < K) hi = mid; else lo = mid + 1;
    }
    const unsigned vstar = lo;

    // count strictly-greater at vstar
    if (tid == 0) shc = 0;
    __syncthreads();
    {
        int cnt = 0;
        for (int p = tid; p < P_; p += 256)
            cnt += (__float_as_uint(lossh[base + p]) > vstar) ? 1 : 0;
        atomicAdd(&shc, cnt);
    }
    __syncthreads();
    const int cgt = shc;
    const int rem = K - cgt;          // >=1: first `rem` equals (by index) are neg
    __syncthreads();

    // stable tie-break by index: contiguous chunks + exclusive prefix
    const int CH = (P_ + 255) / 256;  // 96
    const int st = tid * CH;
    const int en = min(st + CH, P_);
    int ec = 0;
    for (int p = st; p < en; ++p)
        ec += (__float_as_uint(lossh[base + p]) == vstar) ? 1 : 0;
    eqp[tid] = ec;
    __syncthreads();
    if (tid == 0) {
        int run = 0;
        for (int t = 0; t < 256; ++t) { int v = eqp[t]; eqp[t] = run; run += v; }
    }
    __syncthreads();
    int myoff = eqp[tid];
    for (int p = st; p < en; ++p) {
        unsigned bits = __float_as_uint(lossh[base + p]);
        uint8_t n = 0;
        if (bits > vstar) n = 1;
        else if (bits == vstar) { n = (myoff < rem) ? 1 : 0; ++myoff; }
        neg[base + p] = n;
    }
    if (tid == 0) { numpos[b] = npos; numneg[b] = K; }
}

// ---------------------------------------------------------------------------
// Kernel C: streaming CE + GIoU partial sums (deterministic, no global atomics)
// ---------------------------------------------------------------------------
__global__ __launch_bounds__(256)
void mb_loss(const float* __restrict__ loc_data,
             const float* __restrict__ conf_data,
             const float* __restrict__ priors,
             const float* __restrict__ targets,
             const float* __restrict__ cw,
             const uint8_t* __restrict__ conf,
             const uint8_t* __restrict__ bti,
             const uint8_t* __restrict__ neg,
             const float* __restrict__ wscale,
             float* __restrict__ partials)
{
    __shared__ float rl[256], rc[256], rp[256];
    const int tid = threadIdx.x;
    float lc = 0.0f, ll = 0.0f, pc = 0.0f;
    const float4* pr4  = (const float4*)priors;
    const float4* loc4 = (const float4*)loc_data;

    for (int i = blockIdx.x * 256 + tid; i < B_ * P_; i += 256 * NBLK_) {
        __builtin_prefetch(conf_data + (size_t)i * C_ + 4096, 0, 1);
        int b = i / P_;
        int p = i - b * P_;
        int c  = conf[i];
        int ng = neg[i];
        bool pos = (c > 0);
        if (pos || ng) {
            const float* x = conf_data + (size_t)i * C_;
            float m = x[0];
            #pragma unroll
            for (int j = 1; j < C_; ++j) m = fmaxf(m, x[j]);
            float s = 0.0f;
            #pragma unroll
            for (int j = 0; j < C_; ++j) s += expf(x[j] - m);
            float lse = m + logf(s);
            float nll = lse - x[c];
            lc += nll * cw[c];
        }
        if (pos) {
            float4 pr = pr4[p];
            float4 ld = loc4[i];
            // decode(loc_data) * 512
            float cxp = pr.x + ld.x * VAR0_ * pr.z;
            float cyp = pr.y + ld.y * VAR0_ * pr.w;
            float wp  = pr.z * expf(ld.z * VAR1_);
            float hp  = pr.w * expf(ld.w * VAR1_);
            float px1 = (cxp - wp * 0.5f) * 512.0f, py1 = (cyp - hp * 0.5f) * 512.0f;
            float px2 = (cxp + wp * 0.5f) * 512.0f, py2 = (cyp + hp * 0.5f) * 512.0f;
            // matched truth (scaled), encode -> decode roundtrip as in reference
            int o = bti[i];
            const float* t = targets + ((size_t)b * O_ + o) * 5;
            float sc = wscale[b];
            float mx1 = t[0], my1 = t[1], mx2 = t[2] * sc, my2 = t[3] * sc;
            float gcx = ((mx1 + mx2) * 0.5f - pr.x) / (VAR0_ * pr.z);
            float gcy = ((my1 + my2) * 0.5f - pr.y) / (VAR0_ * pr.w);
            float gw  = logf((mx2 - mx1) / pr.z) / VAR1_;
            float gh  = logf((my2 - my1) / pr.w) / VAR1_;
            float cxt = pr.x + gcx * VAR0_ * pr.z;
            float cyt = pr.y + gcy * VAR0_ * pr.w;
            float wt  = pr.z * expf(gw * VAR1_);
            float ht  = pr.w * expf(gh * VAR1_);
            float tx1 = (cxt - wt * 0.5f) * 512.0f, ty1 = (cyt - ht * 0.5f) * 512.0f;
            float tx2 = (cxt + wt * 0.5f) * 512.0f, ty2 = (cyt + ht * 0.5f) * 512.0f;
            // GIoU
            float ap = fmaxf(px2 - px1, 0.0f) * fmaxf(py2 - py1, 0.0f);
            float at = fmaxf(tx2 - tx1, 0.0f) * fmaxf(ty2 - ty1, 0.0f);
            float lx = fmaxf(px1, tx1), lyv = fmaxf(py1, ty1);
            float rx = fminf(px2, tx2), ry  = fminf(py2, ty2);
            float iw = fmaxf(rx - lx, 0.0f), ih = fmaxf(ry - lyv, 0.0f);
            float inter = iw * ih;
            float uni = ap + at - inter;
            float iou = inter / (uni + 1e-7f);
            float c1x = fminf(px1, tx1), c1y = fminf(py1, ty1);
            float c2x = fmaxf(px2, tx2), c2y = fmaxf(py2, ty2);
            float cwd = fmaxf(c2x - c1x, 0.0f), chd = fmaxf(c2y - c1y, 0.0f);
            float ac = cwd * chd;
            float giou = iou - (ac - uni) / (ac + 1e-7f);
            ll += 1.0f - giou;
            pc += 1.0f;
        }
    }
    rl[tid] = ll; rc[tid] = lc; rp[tid] = pc;
    __syncthreads();
    for (int s = 128; s > 0; s >>= 1) {
        if (tid < s) {
            rl[tid] += rl[tid + s];
            rc[tid] += rc[tid + s];
            rp[tid] += rp[tid + s];
        }
        __syncthreads();
    }
    if (tid == 0) {
        partials[blockIdx.x * 3 + 0] = rc[0];
        partials[blockIdx.x * 3 + 1] = rl[0];
        partials[blockIdx.x * 3 + 2] = rp[0];
    }
}

// ---------------------------------------------------------------------------
// Kernel D: final fixed-order reduction
// ---------------------------------------------------------------------------
__global__ void mb_final(const float* __restrict__ partials,
                         const int* __restrict__ numpos,
                         const int* __restrict__ numneg,
                         float* __restrict__ out)
{
    if (threadIdx.x == 0 && blockIdx.x == 0) {
        float lc = 0.0f, ll = 0.0f, pc = 0.0f;
        for (int i = 0; i < NBLK_; ++i) {
            lc += partials[i * 3 + 0];
            ll += partials[i * 3 + 1];
            pc += partials[i * 3 + 2];
        }
        long long np = 0, nn = 0;
        for (int b = 0; b < B_; ++b) { np += numpos[b]; nn += numneg[b]; }
        float N = (float)(np + nn) * 0.5f;
        out[0] = ll / pc;   // loss_l
        out[1] = lc / N;    // loss_c
    }
}

// ---------------------------------------------------------------------------
extern "C" void kernel_launch(void* const* d_in, const int* in_sizes, int n_in,
                              void* d_out, int out_size, void* d_ws, size_t ws_size,
                              hipStream_t stream)
{
    const float* loc_data  = (const float*)d_in[0];
    const float* conf_data = (const float*)d_in[1];
    const float* priors    = (const float*)d_in[2];
    const float* targets   = (const float*)d_in[3];
    const float* cw        = (const float*)d_in[4];
    float* out = (float*)d_out;

    uint8_t* ws = (uint8_t*)d_ws;
    const size_t NP = (size_t)B_ * P_;
    size_t off = 0;
    uint8_t* conf = ws + off; off += NP;
    uint8_t* bti  = ws + off; off += NP;
    uint8_t* neg  = ws + off; off += NP;
    off = (off + 15) & ~(size_t)15;
    float* lossh    = (float*)(ws + off); off += NP * 4;
    float* wscale   = (float*)(ws + off); off += B_ * 4;
    int*   numpos   = (int*)(ws + off);   off += B_ * 4;
    int*   numneg   = (int*)(ws + off);   off += B_ * 4;
    float* partials = (float*)(ws + off); off += (size_t)NBLK_ * 3 * 4;

    mb_match<<<dim3(B_), dim3(256), 0, stream>>>(priors, targets, conf, bti, wscale);
    mb_ohem <<<dim3(B_), dim3(256), 0, stream>>>(conf_data, conf, lossh, neg, numpos, numneg);
    mb_loss <<<dim3(NBLK_), dim3(256), 0, stream>>>(loc_data, conf_data, priors, targets,
                                                    cw, conf, bti, neg, wscale, partials);
    mb_final<<<dim3(1), dim3(32), 0, stream>>>(partials, numpos, numneg, out);
}